// GN_block_14388140442033
// MI455X (gfx1250) — compile-verified
//
#include <hip/hip_runtime.h>
#include <hip/hip_bf16.h>

// ---------------------------------------------------------------------------
// GraphNet block on MI455X (gfx1250).
// N=50000 nodes, E=400000 edges, G=64 graphs, D=128.
// Memory-bound (~1 GB HBM traffic vs 56 GFLOP) -> keep fp32, use
// V_WMMA_F32_16X16X4_F32 for all GEMMs (exact fp32 on the matrix path).
// B (weight) fragments are register-resident per block (128 VGPR per matrix,
// wave32 allows up to 1024 VGPRs); A tiles staged in LDS via gfx1250
// GLOBAL_LOAD_ASYNC_TO_LDS_B128 (ASYNCcnt-tracked DMA, no VGPR round-trip).
// ---------------------------------------------------------------------------

typedef float v2f __attribute__((ext_vector_type(2)));
typedef float v8f __attribute__((ext_vector_type(8)));
typedef int   v4i __attribute__((ext_vector_type(4)));

#define DF   128      // feature dim
#define K2   256      // concat dim
#define KST  64       // K2/4 WMMA k-steps
#define LDA  260      // padded LDS row stride (floats): 260%64==4 -> conflict-free b64
#define TM   16       // rows per WMMA tile
#define MT_PER_BLOCK 8

#define AS1 __attribute__((address_space(1)))
#define AS3 __attribute__((address_space(3)))

#if defined(__gfx1250__) && __has_builtin(__builtin_amdgcn_global_load_async_to_lds_b128)
#define HAVE_ASYNC_LDS 1
#else
#define HAVE_ASYNC_LDS 0
#endif

// 64B global -> LDS copy (per staging thread). Async DMA path on gfx1250.
__device__ __forceinline__ void copy64B_g2l(float* lds_dst, const float* g_src) {
#if HAVE_ASYNC_LDS
#pragma unroll
    for (int j = 0; j < 4; ++j)
        __builtin_amdgcn_global_load_async_to_lds_b128(
            (AS1 v4i*)(g_src + 4 * j), (AS3 v4i*)(lds_dst + 4 * j), 0, 0);
#else
#pragma unroll
    for (int j = 0; j < 4; ++j)
        *(float4*)(lds_dst + 4 * j) = *(const float4*)(g_src + 4 * j);
#endif
}

__device__ __forceinline__ void zero64B_l(float* lds_dst) {
    const float4 z = {0.f, 0.f, 0.f, 0.f};
#pragma unroll
    for (int j = 0; j < 4; ++j) *(float4*)(lds_dst + 4 * j) = z;
}

__device__ __forceinline__ void async_wait_all() {
#if HAVE_ASYNC_LDS
#if __has_builtin(__builtin_amdgcn_s_wait_asynccnt)
    __builtin_amdgcn_s_wait_asynccnt(0);
#else
    asm volatile("s_wait_asynccnt 0x0" ::: "memory");
#endif
#endif
}

// Load one wave's 16-col B slice (K=256) into registers: 64 x v2f = 128 VGPRs.
// B frag map (CDNA5 ISA 7.12.2): lane=(K/2)*16+N, vgpr=K%2.
__device__ __forceinline__ void load_B_frags(const float* __restrict__ W,
                                             int col, int half, v2f* Bw) {
#pragma unroll
    for (int k = 0; k < KST; ++k) {
        const int ka = 4 * k + 2 * half;
        v2f b;
        b.x = W[ka * DF + col];
        b.y = W[(ka + 1) * DF + col];
        Bw[k] = b;
    }
}

// 16x16x256 fp32 GEMM tile: 64 chained V_WMMA_F32_16X16X4_F32.
// A from LDS (lane=(K/2)*16+M), B from registers, C/D lane=(M/8)*16+N vgpr=M%8.
__device__ __forceinline__ v8f wmma_regB(const float* Alds, const v2f* Bw,
                                         int lm, int half) {
    v8f acc = {0.f, 0.f, 0.f, 0.f, 0.f, 0.f, 0.f, 0.f};
#pragma unroll
    for (int k = 0; k < KST; ++k) {
        v2f a = *(const v2f*)&Alds[lm * LDA + 4 * k + 2 * half];
        acc = __builtin_amdgcn_wmma_f32_16x16x4_f32(false, a, false, Bw[k],
                                                    (short)0, acc, false, false);
    }
    return acc;
}

// float atomic max via integer atomics (monotonic bit trick, works for +/-)
__device__ __forceinline__ void atomic_max_f32(float* addr, float v) {
    if (v >= 0.0f) atomicMax((int*)addr, __float_as_int(v));
    else           atomicMin((unsigned int*)addr, __float_as_uint(v));
}

// --------------------------- init kernels ----------------------------------
__global__ void init_ws_kernel(unsigned int* agg_bits, unsigned int* cnt,
                               float* gsum, float* gcnt, int N, int G) {
    int i = blockIdx.x * blockDim.x + threadIdx.x;
    if (i < N * DF) agg_bits[i] = 0xFF800000u;   // -inf
    if (i < N)      cnt[i] = 0u;
    if (i < G * DF) gsum[i] = 0.f;
    if (i < G)      gcnt[i] = 0.f;
}

__global__ void graph_count_kernel(const int* __restrict__ batch, float* gcnt, int N) {
    int i = blockIdx.x * blockDim.x + threadIdx.x;
    if (i < N) atomicAdd(&gcnt[batch[i]], 1.0f);
}

// ------------------- fused edge-MLP + message + scatter-max ----------------
__global__ __launch_bounds__(256)
void edge_msg_kernel(const float* __restrict__ x, const int* __restrict__ ei,
                     const float* __restrict__ eattr,
                     const float* __restrict__ edge_W, const float* __restrict__ edge_b,
                     const float* __restrict__ node_W, const float* __restrict__ node_b,
                     float* __restrict__ out_edge, float* agg, unsigned int* cnt,
                     int E) {
    __shared__ __align__(16) float A[TM * LDA];
    __shared__ int srow[TM], scol[TM];

    const int t    = threadIdx.x;
    const int wave = t >> 5, lane = t & 31;
    const int half = lane >> 4, lm = lane & 15;
    const int nb   = wave * 16;               // this wave's N-slice
    const int col  = nb + lm;
    const int r    = t >> 4;                  // staging row 0..15
    const int seg  = t & 15;                  // staging 16-float segment
    const int* rowp = ei;
    const int* colp = ei + E;

    // Register-resident B for both GEMMs, loaded once per block (amortized x8).
    v2f B1[KST], B2[KST];
    load_B_frags(edge_W, col, half, B1);
    load_B_frags(node_W, col, half, B2);
    const float b1 = edge_b[col];
    const float b2 = node_b[col];

    for (int mt = 0; mt < MT_PER_BLOCK; ++mt) {
        const int e0 = (blockIdx.x * MT_PER_BLOCK + mt) * TM;

        if (t < TM)            srow[t]      = (e0 + t      < E) ? rowp[e0 + t]      : -1;
        else if (t < 2 * TM)   scol[t - TM] = (e0 + t - TM < E) ? colp[e0 + t - TM] : -1;
        __syncthreads();

        // ---- stage A1 = [x[row] | edge_attr] (16 x 256) ----
        {
            const int e = e0 + r;
            float* dst = &A[r * LDA + seg * 16];
            if (e < E) {
                const float* src = (seg < 8)
                    ? &x[(long)srow[r] * DF + seg * 16]
                    : &eattr[(long)e * DF + (seg - 8) * 16];
                copy64B_g2l(dst, src);
            } else {
                zero64B_l(dst);
            }
        }
        async_wait_all();
        __syncthreads();

        // ---- phase 1: edge_new = relu(A1 @ edge_W + edge_b) ----
        v8f acc = wmma_regB(A, B1, lm, half);
        float vals[8];
#pragma unroll
        for (int rr = 0; rr < 8; ++rr) vals[rr] = fmaxf(acc[rr] + b1, 0.0f);
        __syncthreads();   // all waves done reading A1 before overwrite

        // write edge_new -> global AND into LDS cols [128..255] (A2 right half)
#pragma unroll
        for (int rr = 0; rr < 8; ++rr) {
            const int m = 8 * half + rr;
            A[m * LDA + DF + nb + lm] = vals[rr];
            const int e = e0 + m;
            if (e < E) out_edge[(long)e * DF + nb + lm] = vals[rr];
        }
        // stage x[col] into LDS cols [0..127] (A2 left half)
        if (seg < 8) {
            const int e = e0 + r;
            float* dst = &A[r * LDA + seg * 16];
            if (e < E) copy64B_g2l(dst, &x[(long)scol[r] * DF + seg * 16]);
            else       zero64B_l(dst);
        }
        async_wait_all();
        __syncthreads();

        // ---- phase 2: m = A2 @ node_W + node_b, scatter-max into agg[row] ----
        v8f acc2 = wmma_regB(A, B2, lm, half);
#pragma unroll
        for (int rr = 0; rr < 8; ++rr) {
            const int m = 8 * half + rr;
            const int e = e0 + m;
            if (e < E)
                atomic_max_f32(&agg[(long)srow[m] * DF + col], acc2[rr] + b2);
        }
        if (t < TM && (e0 + t) < E) atomicAdd(&cnt[srow[t]], 1u);
        __syncthreads();   // before next tile re-stages LDS
    }
}

// --------------------- node MLP + per-graph sum ----------------------------
__global__ __launch_bounds__(256)
void node_kernel(const float* __restrict__ agg, const unsigned int* __restrict__ cnt,
                 const float* __restrict__ glob, const int* __restrict__ batch,
                 const float* __restrict__ W, const float* __restrict__ bias,
                 float* __restrict__ out_x, float* gsum, int N) {
    __shared__ __align__(16) float A[TM * LDA];
    __shared__ int sb[TM];

    const int t    = threadIdx.x;
    const int wave = t >> 5, lane = t & 31;
    const int half = lane >> 4, lm = lane & 15;
    const int nb   = wave * 16;
    const int col  = nb + lm;
    const int r    = t >> 4;
    const int seg  = t & 15;

    v2f Bw[KST];
    load_B_frags(W, col, half, Bw);
    const float bi = bias[col];

    for (int mt = 0; mt < MT_PER_BLOCK; ++mt) {
        const int n0 = (blockIdx.x * MT_PER_BLOCK + mt) * TM;
        if (t < TM) sb[t] = (n0 + t < N) ? batch[n0 + t] : 0;
        __syncthreads();

        // stage A = [masked agg | glob[batch]] (16 x 256)
        {
            const int node = n0 + r;
            float* dst = &A[r * LDA + seg * 16];
            if (node < N) {
                if (seg < 8) {
                    if (cnt[node] > 0u)
                        copy64B_g2l(dst, &agg[(long)node * DF + seg * 16]);
                    else
                        zero64B_l(dst);   // torch_scatter empty-segment fill
                } else {
                    copy64B_g2l(dst, &glob[(long)sb[r] * DF + (seg - 8) * 16]);
                }
            } else {
                zero64B_l(dst);
            }
        }
        async_wait_all();
        __syncthreads();

        v8f acc = wmma_regB(A, Bw, lm, half);
#pragma unroll
        for (int rr = 0; rr < 8; ++rr) {
            const int m = 8 * half + rr;
            const int node = n0 + m;
            if (node < N) {
                const float v = fmaxf(acc[rr] + bi, 0.0f);
                out_x[(long)node * DF + col] = v;
                atomicAdd(&gsum[(long)sb[m] * DF + col], v);
            }
        }
        __syncthreads();
    }
}

// --------------------------- global MLP ------------------------------------
__global__ __launch_bounds__(256)
void glob_kernel(const float* __restrict__ glob, const float* __restrict__ gsum,
                 const float* __restrict__ gcnt,
                 const float* __restrict__ W, const float* __restrict__ bias,
                 float* __restrict__ out_g, int G) {
    __shared__ __align__(16) float A[TM * LDA];
    const int t    = threadIdx.x;
    const int wave = t >> 5, lane = t & 31;
    const int half = lane >> 4, lm = lane & 15;
    const int nb   = wave * 16;
    const int col  = nb + lm;
    const int r    = t >> 4;
    const int seg  = t & 15;

    v2f Bw[KST];
    load_B_frags(W, col, half, Bw);
    const float bi = bias[col];

    const int ntiles = (G + TM - 1) / TM;
    for (int mt = 0; mt < ntiles; ++mt) {
        const int g0 = mt * TM;
        {
            const int g = g0 + r;
            float* dst = &A[r * LDA + seg * 16];
            if (g < G) {
                if (seg < 8) {
                    copy64B_g2l(dst, &glob[(long)g * DF + seg * 16]);
                } else {
                    const float inv = 1.0f / fmaxf(gcnt[g], 1.0f);
                    const float4* src = (const float4*)&gsum[(long)g * DF + (seg - 8) * 16];
#pragma unroll
                    for (int j = 0; j < 4; ++j) {
                        float4 v = src[j];
                        v.x *= inv; v.y *= inv; v.z *= inv; v.w *= inv;
                        *(float4*)(dst + 4 * j) = v;
                    }
                }
            } else {
                zero64B_l(dst);
            }
        }
        async_wait_all();
        __syncthreads();

        v8f acc = wmma_regB(A, Bw, lm, half);
#pragma unroll
        for (int rr = 0; rr < 8; ++rr) {
            const int m = 8 * half + rr;
            const int g = g0 + m;
            if (g < G) out_g[(long)g * DF + col] = fmaxf(acc[rr] + bi, 0.0f);
        }
        __syncthreads();
    }
}

// --------------------------- host launcher ---------------------------------
extern "C" void kernel_launch(void* const* d_in, const int* in_sizes, int n_in,
                              void* d_out, int out_size, void* d_ws, size_t ws_size,
                              hipStream_t stream) {
    const float* x       = (const float*)d_in[0];
    const int*   ei      = (const int*)  d_in[1];
    const float* eattr   = (const float*)d_in[2];
    const float* glob    = (const float*)d_in[3];
    const int*   batch   = (const int*)  d_in[4];
    const float* edge_W  = (const float*)d_in[5];
    const float* edge_b  = (const float*)d_in[6];
    const float* node_W  = (const float*)d_in[7];
    const float* node_b  = (const float*)d_in[8];
    const float* node2_W = (const float*)d_in[9];
    const float* node2_b = (const float*)d_in[10];
    const float* glob_W  = (const float*)d_in[11];
    const float* glob_b  = (const float*)d_in[12];

    const int N = in_sizes[0] / DF;
    const int E = in_sizes[2] / DF;
    const int G = in_sizes[3] / DF;

    // output layout: x_new [N*D] | edge_new [E*D] | glob_new [G*D]
    float* out_x = (float*)d_out;
    float* out_e = out_x + (size_t)N * DF;
    float* out_g = out_e + (size_t)E * DF;

    // workspace: agg [N*D] f32 | cnt [N] u32 | gsum [G*D] f32 | gcnt [G] f32
    float*        agg  = (float*)d_ws;
    unsigned int* cnt  = (unsigned int*)(agg + (size_t)N * DF);
    float*        gsum = (float*)(cnt + N);
    float*        gcnt = gsum + (size_t)G * DF;

    const int T = 256;
    const int ROWS_PER_BLOCK = TM * MT_PER_BLOCK;

    init_ws_kernel<<<(N * DF + T - 1) / T, T, 0, stream>>>(
        (unsigned int*)agg, cnt, gsum, gcnt, N, G);
    graph_count_kernel<<<(N + T - 1) / T, T, 0, stream>>>(batch, gcnt, N);

    edge_msg_kernel<<<(E + ROWS_PER_BLOCK - 1) / ROWS_PER_BLOCK, T, 0, stream>>>(
        x, ei, eattr, edge_W, edge_b, node_W, node_b, out_e, agg, cnt, E);

    node_kernel<<<(N + ROWS_PER_BLOCK - 1) / ROWS_PER_BLOCK, T, 0, stream>>>(
        agg, cnt, glob, batch, node2_W, node2_b, out_x, gsum, N);

    glob_kernel<<<1, T, 0, stream>>>(glob, gsum, gcnt, glob_W, glob_b, out_g, G);
}